// GRAND_4406636446400
// MI455X (gfx1250) — compile-verified
//
#include <hip/hip_runtime.h>
#include <math.h>

#define IN_DIM 128
#define HID 64
#define OUT_DIM 40
#define OUT_PAD 48
#define EPS_F 0.1f
#define NEG_SLOPE 0.2f

typedef float v2f __attribute__((ext_vector_type(2)));
typedef float v8f __attribute__((ext_vector_type(8)));

// ---------------------------------------------------------------------------
// WMMA f32 16x16x4, wave32 layouts per CDNA5 ISA 7.12.2:
//   A (16x4):  lanes 0-15: row M=lane,    v0=K+0, v1=K+1
//              lanes 16-31: row M=lane-16, v0=K+2, v1=K+3
//   B (4x16):  lanes 0-15: col N=lane,    v0=K+0, v1=K+1
//              lanes 16-31: col N=lane-16, v0=K+2, v1=K+3
//   C/D (16x16): VGPR r -> row r (+8 for lanes>=16), col = lane&15
// B is staged transposed in LDS ([col][K]) so each fragment is one ds_load_b64.
// ---------------------------------------------------------------------------

__device__ __forceinline__ void atomicMaxFloat(float* addr, float val) {
    if (val >= 0.0f)
        atomicMax((int*)addr, __float_as_int(val));
    else
        atomicMin((unsigned int*)addr, __float_as_uint(val));
}

// H[N,64] = X[N,Kdim] @ W[Kdim,64] (+bias). If es!=null also emits
// es[n]=H[n,:].att_src, ed[n]=H[n,:].att_dst via in-register fragment dots.
__global__ void wmma_gemm64(const float* __restrict__ X,
                            const float* __restrict__ W,
                            const float* __restrict__ bias,   // may be null
                            float* __restrict__ H,
                            float* __restrict__ es,           // may be null
                            float* __restrict__ ed,
                            const float* __restrict__ asrc,
                            const float* __restrict__ adst,
                            int N, int Kdim) {
    __shared__ float ldsW[IN_DIM * HID];          // transposed: [col][Kdim]
    const int tid = (int)threadIdx.x;
    for (int i = tid; i < Kdim * HID; i += (int)blockDim.x) {
        int k = i >> 6;                           // i / 64  (coalesced global read)
        int c = i & 63;
        ldsW[c * Kdim + k] = W[(size_t)k * HID + c];
    }
    __syncthreads();

    const int wave = (int)((blockIdx.x * blockDim.x + tid) >> 5);
    const int lane = tid & 31;
    const int nRowTiles = (N + 15) >> 4;
    if (wave >= nRowTiles) return;                // uniform per wave
    const int row0  = wave << 4;
    const int mrow  = lane & 15;
    const int hi    = lane >> 4;
    const int koff  = hi * 2;
    const int arow  = min(row0 + mrow, N - 1);    // clamp: no exec churn

    v8f c0 = {}, c1 = {}, c2 = {}, c3 = {};
    const float* __restrict__ xrow = X + (size_t)arow * Kdim + koff;
    for (int k = 0; k < Kdim; k += 4) {
        float2 av = *(const float2*)(xrow + k);
        v2f a; a.x = av.x; a.y = av.y;
        const float* lp = ldsW + (k + koff);
        float2 f0 = *(const float2*)(lp + (size_t)(mrow)      * Kdim);
        float2 f1 = *(const float2*)(lp + (size_t)(mrow + 16) * Kdim);
        float2 f2 = *(const float2*)(lp + (size_t)(mrow + 32) * Kdim);
        float2 f3 = *(const float2*)(lp + (size_t)(mrow + 48) * Kdim);
        v2f b0, b1, b2, b3;
        b0.x = f0.x; b0.y = f0.y;  b1.x = f1.x; b1.y = f1.y;
        b2.x = f2.x; b2.y = f2.y;  b3.x = f3.x; b3.y = f3.y;
        c0 = __builtin_amdgcn_wmma_f32_16x16x4_f32(false, a, false, b0, (short)0, c0, false, false);
        c1 = __builtin_amdgcn_wmma_f32_16x16x4_f32(false, a, false, b1, (short)0, c1, false, false);
        c2 = __builtin_amdgcn_wmma_f32_16x16x4_f32(false, a, false, b2, (short)0, c2, false, false);
        c3 = __builtin_amdgcn_wmma_f32_16x16x4_f32(false, a, false, b3, (short)0, c3, false, false);
    }

    const int rbase = hi * 8;
    const int ccol  = mrow;
    const float bb0 = bias ? bias[ccol]      : 0.f;
    const float bb1 = bias ? bias[ccol + 16] : 0.f;
    const float bb2 = bias ? bias[ccol + 32] : 0.f;
    const float bb3 = bias ? bias[ccol + 48] : 0.f;
    #pragma unroll
    for (int r = 0; r < 8; ++r) {
        int row = row0 + rbase + r;
        if (row < N) {
            float* hp = H + (size_t)row * HID;
            hp[ccol]      = c0[r] + bb0;
            hp[ccol + 16] = c1[r] + bb1;
            hp[ccol + 32] = c2[r] + bb2;
            hp[ccol + 48] = c3[r] + bb3;
        }
    }

    if (es) {   // fused attention dots: es = z.att_src, ed = z.att_dst
        const float a0 = asrc[ccol], a1 = asrc[ccol + 16],
                    a2 = asrc[ccol + 32], a3 = asrc[ccol + 48];
        const float d0 = adst[ccol], d1 = adst[ccol + 16],
                    d2 = adst[ccol + 32], d3 = adst[ccol + 48];
        #pragma unroll
        for (int r = 0; r < 8; ++r) {
            float vs = c0[r]*a0 + c1[r]*a1 + c2[r]*a2 + c3[r]*a3;
            float vd = c0[r]*d0 + c1[r]*d1 + c2[r]*d2 + c3[r]*d3;
            #pragma unroll
            for (int off = 1; off < 16; off <<= 1) {  // reduce within 16-lane group
                vs += __shfl_xor(vs, off, 32);
                vd += __shfl_xor(vd, off, 32);
            }
            int row = row0 + rbase + r;
            if (ccol == 0 && row < N) { es[row] = vs; ed[row] = vd; }
        }
    }
}

// OUT[N,40] = tanh(H[N,64]) @ Wo[64,40] + bo  (cols padded to 48 in LDS)
__global__ void wmma_out_gemm(const float* __restrict__ H,
                              const float* __restrict__ Wo,
                              const float* __restrict__ bo,
                              float* __restrict__ OUT, int N) {
    __shared__ float ldsW[OUT_PAD * HID];         // transposed: [col][64]
    const int tid = (int)threadIdx.x;
    for (int i = tid; i < OUT_PAD * HID; i += (int)blockDim.x) {
        int c = i >> 6;
        int k = i & 63;
        ldsW[c * HID + k] = (c < OUT_DIM) ? Wo[(size_t)k * OUT_DIM + c] : 0.f;
    }
    __syncthreads();

    const int wave = (int)((blockIdx.x * blockDim.x + tid) >> 5);
    const int lane = tid & 31;
    const int nRowTiles = (N + 15) >> 4;
    if (wave >= nRowTiles) return;
    const int row0 = wave << 4;
    const int mrow = lane & 15;
    const int hi   = lane >> 4;
    const int koff = hi * 2;
    const int arow = min(row0 + mrow, N - 1);

    v8f c0 = {}, c1 = {}, c2 = {};
    const float* __restrict__ hrow = H + (size_t)arow * HID + koff;
    for (int k = 0; k < HID; k += 4) {
        float2 av = *(const float2*)(hrow + k);
        v2f a; a.x = tanhf(av.x); a.y = tanhf(av.y);
        const float* lp = ldsW + (k + koff);
        float2 f0 = *(const float2*)(lp + (size_t)(mrow)      * HID);
        float2 f1 = *(const float2*)(lp + (size_t)(mrow + 16) * HID);
        float2 f2 = *(const float2*)(lp + (size_t)(mrow + 32) * HID);
        v2f b0, b1, b2;
        b0.x = f0.x; b0.y = f0.y;  b1.x = f1.x; b1.y = f1.y;
        b2.x = f2.x; b2.y = f2.y;
        c0 = __builtin_amdgcn_wmma_f32_16x16x4_f32(false, a, false, b0, (short)0, c0, false, false);
        c1 = __builtin_amdgcn_wmma_f32_16x16x4_f32(false, a, false, b1, (short)0, c1, false, false);
        c2 = __builtin_amdgcn_wmma_f32_16x16x4_f32(false, a, false, b2, (short)0, c2, false, false);
    }
    const int rbase = hi * 8;
    const int ccol  = mrow;
    #pragma unroll
    for (int r = 0; r < 8; ++r) {
        int row = row0 + rbase + r;
        if (row < N) {
            float* op = OUT + (size_t)row * OUT_DIM;
            op[ccol] = c0[r] + bo[ccol];
            if (ccol + 16 < OUT_DIM) op[ccol + 16] = c1[r] + bo[ccol + 16];
            if (ccol + 32 < OUT_DIM) op[ccol + 32] = c2[r] + bo[ccol + 32];
        }
    }
}

// agg[N*64]=0 (float4), m=-3e38, denom=0
__global__ void seg_init(float4* __restrict__ agg4, float* __restrict__ m,
                         float* __restrict__ denom, int N) {
    size_t t = (size_t)blockIdx.x * blockDim.x + threadIdx.x;
    size_t tot4 = (size_t)N * (HID / 4);
    if (t < tot4) agg4[t] = make_float4(0.f, 0.f, 0.f, 0.f);
    if (t < (size_t)N) { m[t] = -3.0e38f; denom[t] = 0.f; }
}

// pass 1: e = leaky_relu(es[src]+ed[dst]); atomic max per dst
__global__ void edge_pass1(const long long* __restrict__ ei,
                           const float* __restrict__ es, const float* __restrict__ ed,
                           float* __restrict__ ebuf, float* __restrict__ m,
                           int E, int N) {
    long long t = (long long)blockIdx.x * blockDim.x + threadIdx.x;
    if (t >= (long long)E + N) return;
    int s, d;
    if (t < E) { s = (int)ei[t]; d = (int)ei[(long long)E + t]; }
    else       { s = d = (int)(t - E); }
    float e = es[s] + ed[d];
    e = (e > 0.f) ? e : NEG_SLOPE * e;
    ebuf[t] = e;
    atomicMaxFloat(m + d, e);
}

// pass 2: w = exp(e - m[dst]); denom[dst] += w
__global__ void edge_pass2(const long long* __restrict__ ei,
                           float* __restrict__ ebuf,
                           const float* __restrict__ m, float* __restrict__ denom,
                           int E, int N) {
    long long t = (long long)blockIdx.x * blockDim.x + threadIdx.x;
    if (t >= (long long)E + N) return;
    int d = (t < E) ? (int)ei[(long long)E + t] : (int)(t - E);
    float w = __expf(ebuf[t] - m[d]);
    ebuf[t] = w;
    atomicAdd(denom + d, w);
}

// pass 3: agg[dst,:] += (w/denom[dst]) * z[src,:]; 64 threads/edge, coalesced
__global__ void edge_scatter(const long long* __restrict__ ei,
                             const float* __restrict__ ebuf,
                             const float* __restrict__ denom,
                             const float* __restrict__ Z,
                             float* __restrict__ agg, int E, int N) {
    long long tid = (long long)blockIdx.x * blockDim.x + threadIdx.x;
    long long edge = tid >> 6;
    int k = (int)(tid & 63);
    if (edge >= (long long)E + N) return;
    int s, d;
    if (edge < E) { s = (int)ei[edge]; d = (int)ei[(long long)E + edge]; }
    else          { s = d = (int)(edge - E); }
    float coef = ebuf[edge] / denom[d];
    atomicAdd(agg + (size_t)d * HID + k, coef * Z[(size_t)s * HID + k]);
}

// RK4 combine (float4): f = alpha*(agg + b_gat - y_in); per-stage update
__global__ void rk4_combine(const float4* __restrict__ agg4,
                            const float4* __restrict__ yin4,
                            float4* __restrict__ h4, float4* __restrict__ y4,
                            float4* __restrict__ acc4,
                            const float* __restrict__ bgat,
                            const float* __restrict__ alpha_p,
                            int N, int stage) {
    size_t t = (size_t)blockIdx.x * blockDim.x + threadIdx.x;
    size_t tot4 = (size_t)N * (HID / 4);
    if (t >= tot4) return;
    int kq = (int)(t & (HID / 4 - 1)) * 4;
    float alpha = 1.0f / (1.0f + __expf(-alpha_p[0]));
    float4 ag = agg4[t];
    float4 yi = yin4[t];
    const float4 bg = *(const float4*)(bgat + kq);
    float4 f;
    f.x = alpha * (ag.x + bg.x - yi.x);
    f.y = alpha * (ag.y + bg.y - yi.y);
    f.z = alpha * (ag.z + bg.z - yi.z);
    f.w = alpha * (ag.w + bg.w - yi.w);
    if (stage == 3) {
        float4 ac = acc4[t];
        float4 hv = h4[t];
        hv.x += (EPS_F / 6.0f) * (ac.x + f.x);
        hv.y += (EPS_F / 6.0f) * (ac.y + f.y);
        hv.z += (EPS_F / 6.0f) * (ac.z + f.z);
        hv.w += (EPS_F / 6.0f) * (ac.w + f.w);
        h4[t] = hv;
    } else {
        float step = (stage == 2) ? EPS_F : 0.5f * EPS_F;
        float4 hv = h4[t];
        float4 yn;
        yn.x = hv.x + step * f.x;  yn.y = hv.y + step * f.y;
        yn.z = hv.z + step * f.z;  yn.w = hv.w + step * f.w;
        y4[t] = yn;
        if (stage == 0) {
            acc4[t] = f;
        } else {
            float4 ac = acc4[t];
            ac.x += 2.0f * f.x;  ac.y += 2.0f * f.y;
            ac.z += 2.0f * f.z;  ac.w += 2.0f * f.w;
            acc4[t] = ac;
        }
    }
}

extern "C" void kernel_launch(void* const* d_in, const int* in_sizes, int n_in,
                              void* d_out, int out_size, void* d_ws, size_t ws_size,
                              hipStream_t stream) {
    const float*     x       = (const float*)d_in[0];
    const long long* ei      = (const long long*)d_in[1];  // int64 [2,E]
    const float*     W_emb   = (const float*)d_in[2];
    const float*     b_emb   = (const float*)d_in[3];
    const float*     W_gat   = (const float*)d_in[4];
    const float*     att_src = (const float*)d_in[5];
    const float*     att_dst = (const float*)d_in[6];
    const float*     b_gat   = (const float*)d_in[7];
    const float*     alpha_p = (const float*)d_in[8];
    const float*     W_out   = (const float*)d_in[9];
    const float*     b_out   = (const float*)d_in[10];
    float*           out     = (float*)d_out;

    const int N = in_sizes[0] / IN_DIM;
    const int E = in_sizes[1] / 2;
    const long long EN = (long long)E + N;

    // workspace layout (floats)
    float* p = (float*)d_ws;
    float* h     = p; p += (size_t)N * HID;
    float* y     = p; p += (size_t)N * HID;
    float* z     = p; p += (size_t)N * HID;
    float* agg   = p; p += (size_t)N * HID;
    float* acc   = p; p += (size_t)N * HID;
    float* es    = p; p += N;
    float* ed    = p; p += N;
    float* m     = p; p += N;
    float* denom = p; p += N;
    float* ebuf  = p; p += EN;

    const int TB = 256;
    const int wavesPerBlk = TB / 32;
    const int rowTiles = (N + 15) / 16;
    const int gemmBlks = (rowTiles + wavesPerBlk - 1) / wavesPerBlk;
    const int nodeHid4Blks = (int)(((size_t)N * (HID / 4) + TB - 1) / TB);
    const int edgeBlks = (int)((EN + TB - 1) / TB);
    const int scatBlks = (int)((EN * HID + TB - 1) / TB);

    // h = x @ W_emb + b_emb
    wmma_gemm64<<<gemmBlks, TB, 0, stream>>>(x, W_emb, b_emb, h,
                                             nullptr, nullptr, nullptr, nullptr,
                                             N, IN_DIM);

    for (int stage = 0; stage < 4; ++stage) {
        const float* yin = (stage == 0) ? h : y;
        // z = yin @ W_gat ; fused es/ed dots
        wmma_gemm64<<<gemmBlks, TB, 0, stream>>>(yin, W_gat, nullptr, z,
                                                 es, ed, att_src, att_dst,
                                                 N, HID);
        seg_init<<<nodeHid4Blks, TB, 0, stream>>>((float4*)agg, m, denom, N);
        edge_pass1<<<edgeBlks, TB, 0, stream>>>(ei, es, ed, ebuf, m, E, N);
        edge_pass2<<<edgeBlks, TB, 0, stream>>>(ei, ebuf, m, denom, E, N);
        edge_scatter<<<scatBlks, TB, 0, stream>>>(ei, ebuf, denom, z, agg, E, N);
        rk4_combine<<<nodeHid4Blks, TB, 0, stream>>>((const float4*)agg,
                                                     (const float4*)yin,
                                                     (float4*)h, (float4*)y,
                                                     (float4*)acc,
                                                     b_gat, alpha_p, N, stage);
    }

    // out = tanh(h) @ W_out + b_out
    wmma_out_gemm<<<gemmBlks, TB, 0, stream>>>(h, W_out, b_out, out, N);
}